// Net_84052509983223
// MI455X (gfx1250) — compile-verified
//
#include <hip/hip_runtime.h>
#include <math.h>

typedef __attribute__((ext_vector_type(2))) float v2f;
typedef __attribute__((ext_vector_type(8))) float v8f;

#define BATCH 512
#define CH 3
#define NPIX 128
#define NN (NPIX * NPIX)          // 16384
#define LAYERS 3
#define TILE 32
#define HALO 3
#define RIN (TILE + 2 * HALO)     // 38
#define TILES_PER_IMG 16
#define H_ELEMS (BATCH * CH * NN) // 25165824

__device__ __forceinline__ float block_reduce_sum(float v, float* red) {
  const int tid = threadIdx.x;
  red[tid] = v;
  __syncthreads();
#pragma unroll
  for (int s = 128; s > 0; s >>= 1) {
    if (tid < s) red[tid] += red[tid + s];
    __syncthreads();
  }
  return red[0];
}

// ---------------------------------------------------------------------------
// Fused 3-layer flow.  Per layer: actnorm is folded into the WMMA operands
// (B-fragment scaled by exp(act_ls) per input channel; actnorm bias becomes a
// per-out-channel constant loaded into the WMMA C accumulator), circular 3x3
// conv runs as implicit GEMM on V_WMMA_F32_16X16X4_F32 (M=16 px, K=27->28,
// N=3->16), then an elementwise epilogue does exp(cls)*y + bias and the slog
// gate, accumulating the per-tile data-dependent logdet part.
// ---------------------------------------------------------------------------
__global__ __launch_bounds__(256) void flow_main_kernel(
    const float* __restrict__ x,
    const float* __restrict__ act_bias,
    const float* __restrict__ act_ls,
    const float* __restrict__ kern,
    const float* __restrict__ conv_bias,
    const float* __restrict__ conv_ls,
    const float* __restrict__ slog_la,
    float* __restrict__ out_h,
    float* __restrict__ ws_tile)
{
  __shared__ float bufA[CH * RIN * RIN];   // 38x38x3
  __shared__ float bufB[CH * 36 * 36];     // 36x36x3 (layer-0 output region)
  __shared__ float kwS[LAYERS * 81];
  __shared__ float ascS[9], abS[9], alS[9], ralS[9], csumS[9];
  __shared__ float red[256];

  const int tid = threadIdx.x;
  const int b   = blockIdx.x >> 4;
  const int t   = blockIdx.x & 15;
  const int ty0 = (t >> 2) * TILE;
  const int tx0 = (t & 3) * TILE;

  if (tid < 9) {
    ascS[tid]  = expf(act_ls[tid]);
    abS[tid]   = act_bias[tid];
    alS[tid]   = expf(slog_la[tid]);
    ralS[tid]  = expf(-slog_la[tid]);   // 1/alpha
  }
  for (int i = tid; i < LAYERS * 81; i += 256) kwS[i] = kern[i];
  __syncthreads();

  // Per-layer, per-out-channel actnorm-bias constant: sum_i b_i * sum_t K[o,i,t]
  if (tid < 9) {
    const int l = tid / 3, o = tid - 3 * l;
    float s = 0.f;
#pragma unroll
    for (int i = 0; i < 3; ++i) {
      float ks = 0.f;
#pragma unroll
      for (int tt = 0; tt < 9; ++tt) ks += kwS[l * 81 + o * 27 + i * 9 + tt];
      s += abS[l * 3 + i] * ks;
    }
    csumS[tid] = s;
  }

  // Pure copy of the wrapped 38x38x3 input region (actnorm folded into GEMM).
  const float* xb = x + (size_t)b * CH * NN;
  for (int i = tid; i < CH * RIN * RIN; i += 256) {
    int c  = i / (RIN * RIN);
    int r  = i - c * (RIN * RIN);
    int ly = r / RIN, lx = r - ly * RIN;
    int gy = (ty0 + ly - HALO) & (NPIX - 1);
    int gx = (tx0 + lx - HALO) & (NPIX - 1);
    bufA[i] = xb[c * NN + gy * NPIX + gx];
  }
  __syncthreads();

  const int lane   = tid & 31;
  const int wave   = tid >> 5;
  const int lane16 = lane & 15;
  const int hi     = lane >> 4;

  float acc = 0.0f;

#pragma unroll
  for (int l = 0; l < LAYERS; ++l) {
    const int Sin  = RIN - 2 * l;
    const int Sout = Sin - 2;
    const int P    = Sout * Sout;
    const int T    = (P + 15) >> 4;
    float* inb  = (l == 1) ? bufB : bufA;
    float* outb = (l == 1) ? bufA : bufB;

    // B fragments (28x16, zero padded), actnorm input scale folded in.
    // Layout mirrors C/D: v0 rows {4q,4q+2} across lane halves, v1 {4q+1,4q+3}.
    v2f bf[7];
#pragma unroll
    for (int q = 0; q < 7; ++q) {
      const int kk = 4 * q + 2 * hi;
      const int n  = lane16;
      float bx = 0.f, by = 0.f;
      if (n < 3) {
        if (kk < 27)     bx = kwS[l * 81 + n * 27 + kk] * ascS[l * 3 + kk / 9];
        if (kk + 1 < 27) by = kwS[l * 81 + n * 27 + kk + 1] * ascS[l * 3 + (kk + 1) / 9];
      }
      bf[q].x = bx;
      bf[q].y = by;
    }
    const float cinit = (lane16 < 3) ? csumS[l * 3 + lane16] : 0.f;

    // Implicit-GEMM conv: M = 16 pixels, K = 27 (im2col), N = 3 out channels.
    for (int tile = wave; tile < T; tile += 8) {
      const int m  = tile * 16 + lane16;     // A row (valid if m < P)
      const int ly = m / Sout;
      const int lx = m - ly * Sout;
      v8f c = {cinit, cinit, cinit, cinit, cinit, cinit, cinit, cinit};
#pragma unroll
      for (int q = 0; q < 7; ++q) {
        const int k0 = 4 * q + 2 * hi;
        v2f a;
        {
          int ci = k0 / 9, tt = k0 - ci * 9;
          int dy = tt / 3, dx = tt - dy * 3;
          float v = inb[ci * Sin * Sin + (ly + dy) * Sin + (lx + dx)];
          a.x = (k0 < 27) ? v : 0.f;
        }
        {
          int k1 = k0 + 1;
          int ci = k1 / 9, tt = k1 - ci * 9;
          int dy = tt / 3, dx = tt - dy * 3;
          float v = inb[ci * Sin * Sin + (ly + dy) * Sin + (lx + dx)];
          a.y = (k1 < 27) ? v : 0.f;
        }
        c = __builtin_amdgcn_wmma_f32_16x16x4_f32(false, a, false, bf[q],
                                                  (short)0, c, false, false);
      }
      // Scatter D: lane holds column n = lane&15; vgpr r holds row r (+8 hi).
      if (lane16 < 3) {
        const int n = lane16;
#pragma unroll
        for (int r = 0; r < 8; ++r) {
          const int mm = tile * 16 + r + hi * 8;
          if (mm < P) outb[n * P + mm] = c[r];
        }
      }
    }
    __syncthreads();

    // Elementwise epilogue: exp(cls)*y + bias, slog gate, logdet partial.
    const int off = 2 - l;  // remaining halo of this layer's output region
    for (int i = tid; i < CH * P; i += 256) {
      const int n  = i / P;
      const int rr = i - n * P;
      const int ly = rr / Sout, lx = rr - ly * Sout;
      const int gy = (ty0 + ly - off) & (NPIX - 1);
      const int gx = (tx0 + lx - off) & (NPIX - 1);
      const int gidx = l * (CH * NN) + n * NN + gy * NPIX + gx;
      const float y  = outb[i];
      const float h2 = __expf(conv_ls[gidx]) * y + conv_bias[gidx];
      const float lg = log1pf(alS[l * 3 + n] * fabsf(h2));
      const float h3 = copysignf(lg * ralS[l * 3 + n], h2);
      // -alpha*|h3| == -log1p(alpha*|h2|); count each global pixel once.
      if ((unsigned)(ly - off) < (unsigned)TILE &&
          (unsigned)(lx - off) < (unsigned)TILE) acc -= lg;
      if (l < 2) {
        outb[i] = h3;   // next layer's actnorm is folded into its GEMM
      } else {
        out_h[(size_t)b * CH * NN + n * NN + gy * NPIX + gx] = h3;
      }
    }
    __syncthreads();
  }

  const float tot = block_reduce_sum(acc, red);
  if (tid == 0) ws_tile[blockIdx.x] = tot;
}

// ---------------------------------------------------------------------------
// Data-independent logdet pieces: FFT spectral log|det| of each 3x3 kernel,
// sum(conv_log_scale), and n^2 * sum(actnorm_log_scale).  256 partials -> ws.
// ---------------------------------------------------------------------------
struct cpx { float r, i; };
__device__ __forceinline__ cpx cmul(cpx a, cpx b) {
  return {a.r * b.r - a.i * b.i, a.r * b.i + a.i * b.r};
}
__device__ __forceinline__ cpx csub(cpx a, cpx b) { return {a.r - b.r, a.i - b.i}; }
__device__ __forceinline__ cpx cadd(cpx a, cpx b) { return {a.r + b.r, a.i + b.i}; }

__global__ __launch_bounds__(256) void flow_const_kernel(
    const float* __restrict__ kern,
    const float* __restrict__ act_ls,
    const float* __restrict__ conv_ls,
    float* __restrict__ ws_part)
{
  __shared__ float red[256];
  const int tid = threadIdx.x;
  float acc = 0.0f;

  if (blockIdx.x < 192) {
    // One (layer, frequency) per thread: 3 * 16384 items over 192 blocks.
    const int item = blockIdx.x * 256 + tid;
    const int l = item / NN;
    const int f = item - l * NN;
    const int u = f >> 7, v = f & 127;
    float re[9] = {0, 0, 0, 0, 0, 0, 0, 0, 0};
    float im[9] = {0, 0, 0, 0, 0, 0, 0, 0, 0};
#pragma unroll
    for (int dy = 0; dy < 3; ++dy) {
#pragma unroll
      for (int dx = 0; dx < 3; ++dx) {
        const float ang = -(6.28318530717958647692f / 128.0f) *
                          (float)(u * dy + v * dx);
        float s, cth;
        sincosf(ang, &s, &cth);
#pragma unroll
        for (int a2 = 0; a2 < 3; ++a2) {
#pragma unroll
          for (int b2 = 0; b2 < 3; ++b2) {
            const float w = kern[l * 81 + a2 * 27 + b2 * 9 + dy * 3 + dx];
            re[a2 * 3 + b2] += w * cth;
            im[a2 * 3 + b2] += w * s;
          }
        }
      }
    }
    cpx m[9];
#pragma unroll
    for (int j = 0; j < 9; ++j) m[j] = {re[j], im[j]};
    const cpx d0 = cmul(m[0], csub(cmul(m[4], m[8]), cmul(m[5], m[7])));
    const cpx d1 = cmul(m[1], csub(cmul(m[3], m[8]), cmul(m[5], m[6])));
    const cpx d2 = cmul(m[2], csub(cmul(m[3], m[7]), cmul(m[4], m[6])));
    const cpx det = cadd(csub(d0, d1), d2);
    acc = 0.5f * logf(det.r * det.r + det.i * det.i);
  } else {
    // Blocks 192..255: sum conv_log_scale (L*C*N*N elements).
    const int base = (blockIdx.x - 192) * 256 + tid;
    for (int j = base; j < LAYERS * CH * NN; j += 64 * 256) acc += conv_ls[j];
    if (blockIdx.x == 192 && tid == 0) {
      float s = 0.f;
      for (int i2 = 0; i2 < 9; ++i2) s += act_ls[i2];
      acc += (float)NN * s;  // (n*n) * sum(actnorm_log_scale), all layers
    }
  }

  const float tot = block_reduce_sum(acc, red);
  if (tid == 0) ws_part[blockIdx.x] = tot;
}

// ---------------------------------------------------------------------------
// Per-batch finalize: scalar const (256 partials) + 16 tile partials.
// Fixed-order tree reduction -> deterministic.
// ---------------------------------------------------------------------------
__global__ __launch_bounds__(256) void flow_finalize_kernel(
    const float* __restrict__ ws_part,
    const float* __restrict__ ws_tile,
    float* __restrict__ out_ld)
{
  __shared__ float red[256];
  const int tid = threadIdx.x;
  float v = ws_part[tid];
  if (tid < TILES_PER_IMG) v += ws_tile[blockIdx.x * TILES_PER_IMG + tid];
  const float tot = block_reduce_sum(v, red);
  if (tid == 0) out_ld[blockIdx.x] = tot;
}

extern "C" void kernel_launch(void* const* d_in, const int* in_sizes, int n_in,
                              void* d_out, int out_size, void* d_ws, size_t ws_size,
                              hipStream_t stream) {
  const float* x    = (const float*)d_in[0];
  const float* anb  = (const float*)d_in[1];  // actnorm_bias      (L,1,C,1,1)
  const float* anls = (const float*)d_in[2];  // actnorm_log_scale (L,1,C,1,1)
  const float* kw   = (const float*)d_in[3];  // conv_kernel       (L,C,C,3,3)
  const float* cb   = (const float*)d_in[4];  // conv_bias         (L,1,C,N,N)
  const float* cls  = (const float*)d_in[5];  // conv_log_scale    (L,1,C,N,N)
  const float* sla  = (const float*)d_in[6];  // slog_log_alpha    (L,1,C,1,1)

  float* out     = (float*)d_out;
  float* ws_part = (float*)d_ws;        // 256 const partials
  float* ws_tile = ws_part + 256;       // 8192 tile partials

  flow_const_kernel<<<256, 256, 0, stream>>>(kw, anls, cls, ws_part);
  flow_main_kernel<<<BATCH * TILES_PER_IMG, 256, 0, stream>>>(
      x, anb, anls, kw, cb, cls, sla, out, ws_tile);
  flow_finalize_kernel<<<BATCH, 256, 0, stream>>>(ws_part, ws_tile,
                                                  out + H_ELEMS);
}